// skipSRNN_ALIF_1Adapt_40046275068556
// MI455X (gfx1250) — compile-verified
//
#include <hip/hip_runtime.h>
#include <math.h>

typedef float v2f __attribute__((ext_vector_type(2)));
typedef float v8f __attribute__((ext_vector_type(8)));
typedef int   v2i __attribute__((ext_vector_type(2)));

#define NB   32      // batch
#define TT   100     // time steps
#define IND  1024    // input dim after pooling (32*32)
#define FC   512
#define OUTD 11
#define MROWS (TT*NB)   // 3200

// ---------------- WMMA helper (fp32 16x16x4) ----------------
__device__ inline v8f wmma_f32(v2f a, v2f b, v8f c) {
  return __builtin_amdgcn_wmma_f32_16x16x4_f32(false, a, false, b, (short)0, c, false, false);
}

// ---------------- async global->LDS copy of 8 bytes ----------------
__device__ inline void async_copy8(const float* gsrc, float* ldst) {
  __builtin_amdgcn_global_load_async_to_lds_b64(
      (__attribute__((address_space(1))) v2i*)gsrc,
      (__attribute__((address_space(3))) v2i*)ldst, 0, 0);
}

// ---------------- init: zero sp ping-pong buffers + barrier ----------------
__global__ void init_kernel(float* sp0, float* sp1, int* bar) {
  int tid = blockIdx.x * blockDim.x + threadIdx.x;
  if (tid < NB * FC) { sp0[tid] = 0.f; sp1[tid] = 0.f; }
  if (tid < 2) bar[tid] = 0;
}

// ---------------- 4x4 max pool, split channels, layout [T*N, 1024] ----------------
__global__ __launch_bounds__(256) void pool_kernel(const float* __restrict__ x,
                                                   float* __restrict__ x0,
                                                   float* __restrict__ x1) {
  int tid = blockIdx.x * blockDim.x + threadIdx.x;   // 2*3200*1024 total
  if (tid >= 2 * MROWS * IND) return;
  int pw = tid & 31;
  int ph = (tid >> 5) & 31;
  int c  = (tid >> 10) & 1;
  int r  = tid >> 11;            // r = t*NB + n
  int t  = r >> 5;
  int n  = r & 31;
  const float* base = x + ((((size_t)(n * TT + t) * 2 + c) * 128 + ph * 4) * 128 + pw * 4);
  float m = -INFINITY;
  #pragma unroll
  for (int i = 0; i < 4; ++i) {
    float4 v = *(const float4*)(base + i * 128);
    m = fmaxf(m, fmaxf(fmaxf(v.x, v.y), fmaxf(v.z, v.w)));
  }
  float* dst = c ? x1 : x0;
  dst[(size_t)r * IND + ph * 32 + pw] = m;
}

// ---------------- batched GEMM: Y = X @ W^T + bias1 (+bias2) -------------
// Block = 8 waves = 64(M) x 128(N) tile. K-chunks of 32 are async-staged into
// LDS in WMMA-fragment order (coalesced global reads, scattered LDS writes),
// double-buffered so copies overlap WMMA. Each wave computes 16x64.
// grid = (N/128, M/64); requires K % 32 == 0.
__global__ __launch_bounds__(256) void gemm_wmma(const float* __restrict__ X,
                                                 const float* __restrict__ W,
                                                 const float* __restrict__ bias1,
                                                 const float* __restrict__ bias2,
                                                 float* __restrict__ Y,
                                                 int M, int K, int N) {
  // one buffer: A frags [4 mt][8 ks][32 lane]{2f} = 2048 floats,
  //             B frags [8 nt][8 ks][32 lane]{2f} = 4096 floats
  __shared__ float lds[2 * 6144];          // 48 KB double-buffered
  int lane = threadIdx.x & 31;
  int wid  = threadIdx.x >> 5;
  int wm = wid >> 1, wn = wid & 1;
  int mbase0 = blockIdx.y * 64;
  int nbase0 = blockIdx.x * 128;
  int mbase = mbase0 + wm * 16;
  int l15 = lane & 15;
  int lh  = lane >> 4;
  int j0 = nbase0 + wn * 64 + l15;

  v8f acc[4];
  #pragma unroll
  for (int i = 0; i < 4; ++i) {
    int j = j0 + 16 * i;
    float b = bias1[j] + (bias2 ? bias2[j] : 0.f);
    #pragma unroll
    for (int v = 0; v < 8; ++v) acc[i][v] = b;
  }

  const int NC = K >> 5;                   // number of 32-wide K chunks

  // async-stage one K-chunk into fragment order; global reads coalesced.
  auto issue = [&](float* buf, int kb) {
    // A tile: 64 rows x 16 pairs = 1024 chunks, 4 rounds of 256 threads
    #pragma unroll
    for (int rd = 0; rd < 4; ++rd) {
      int c = rd * 256 + threadIdx.x;
      int r = c >> 4, p = c & 15;
      int mt = r >> 4, lr = r & 15, ks = p >> 1, ph = p & 1;
      async_copy8(X + (size_t)(mbase0 + r) * K + kb + p * 2,
                  buf + (((mt * 8 + ks) * 32) + ph * 16 + lr) * 2);
    }
    // B tile: 128 rows x 16 pairs = 2048 chunks, 8 rounds
    #pragma unroll
    for (int rd = 0; rd < 8; ++rd) {
      int c = rd * 256 + threadIdx.x;
      int r = c >> 4, p = c & 15;
      int ntile = r >> 4, lr = r & 15, ks = p >> 1, ph = p & 1;
      async_copy8(W + (size_t)(nbase0 + r) * K + kb + p * 2,
                  buf + 2048 + (((ntile * 8 + ks) * 32) + ph * 16 + lr) * 2);
    }
  };

  issue(lds, 0);
  __builtin_amdgcn_s_wait_asynccnt(0);
  __syncthreads();

  for (int cidx = 0; cidx < NC; ++cidx) {
    float* buf = lds + (cidx & 1) * 6144;
    if (cidx + 1 < NC) issue(lds + ((cidx + 1) & 1) * 6144, (cidx + 1) << 5);

    const float* Af = buf + (wm * 8) * 64 + lane * 2;
    const float* Bf = buf + 2048 + (wn * 4 * 8) * 64 + lane * 2;
    #pragma unroll
    for (int ks = 0; ks < 8; ++ks) {
      v2f a = *(const v2f*)(Af + ks * 64);
      #pragma unroll
      for (int i = 0; i < 4; ++i) {
        v2f b = *(const v2f*)(Bf + (i * 8 + ks) * 64);
        acc[i] = wmma_f32(a, b, acc[i]);
      }
    }
    if (cidx + 1 < NC) {
      __builtin_amdgcn_s_wait_asynccnt(0);
      __syncthreads();
    }
  }

  #pragma unroll
  for (int v = 0; v < 8; ++v) {
    int m = mbase + v + lh * 8;
    #pragma unroll
    for (int i = 0; i < 4; ++i)
      Y[(size_t)m * N + j0 + 16 * i] = acc[i][v];
  }
}

// ---------------- layers 1&2: per-element scan over T (beta=0 => Bth=0.1) ----------------
__global__ __launch_bounds__(256) void scan12_kernel(const float* __restrict__ H1,
                                                     const float* __restrict__ H2,
                                                     const int* __restrict__ mask1,
                                                     const int* __restrict__ mask2,
                                                     const float* __restrict__ tau_m1,
                                                     const float* __restrict__ tau_m2,
                                                     float* __restrict__ SPcat) {
  int tid = blockIdx.x * blockDim.x + threadIdx.x;
  if (tid >= 2 * NB * FC) return;
  int j = tid & (FC - 1);
  int n = (tid >> 9) & (NB - 1);
  int L = tid >> 14;
  const float* H    = L ? H2 : H1;
  const int*   mask = L ? mask2 : mask1;
  float tm = (L ? tau_m2 : tau_m1)[j];
  float alpha = expf(-1.f / tm);
  float onea  = 1.f - alpha;
  float mem = 0.f, spike = 0.f;
  for (int t = 0; t < TT; ++t) {
    float h  = H[(size_t)(t * NB + n) * FC + j];
    float nm = mem * alpha + onea * h - 0.1f * spike;
    int mk = mask[j * TT + t];
    if (mk) mem = nm;
    spike = (mk && (mem - 0.1f > 0.f)) ? 1.f : 0.f;
    SPcat[(size_t)(t * NB + n) * (2 * FC) + L * FC + j] = spike;
  }
}

// ---------------- grid barrier: HW cluster barrier if clustered, else L2 spin ----------------
__device__ inline int read_cluster_id() {
  int v;
  asm volatile("s_getreg_b32 %0, hwreg(HW_REG_IB_STS2, 6, 4)" : "=s"(v));
  return v;
}

__device__ inline void grid_barrier(int* cnt, int* gen, int step, int clustered) {
  __builtin_amdgcn_fence(__ATOMIC_RELEASE, "agent");
  __syncthreads();
  if (clustered) {
    if (threadIdx.x < 32) asm volatile("s_barrier_signal -3");
    asm volatile("s_barrier_wait -3");
  } else {
    if (threadIdx.x == 0) {
      int arrive = atomicAdd(cnt, 1);
      if (arrive == (int)gridDim.x - 1) {
        atomicExch(cnt, 0);
        atomicAdd(gen, 1);
      } else {
        while (__hip_atomic_load(gen, __ATOMIC_ACQUIRE, __HIP_MEMORY_SCOPE_AGENT) <= step) {
          __builtin_amdgcn_s_sleep(1);
        }
      }
    }
  }
  __syncthreads();
  __builtin_amdgcn_fence(__ATOMIC_ACQUIRE, "agent");
}

// ---------------- persistent recurrent layer 3 ----------------
__global__ __launch_bounds__(128) void recurrent_kernel(const float* __restrict__ G3,
                                                        const float* __restrict__ Wh,
                                                        const float* __restrict__ tau_adp3,
                                                        const float* __restrict__ tau_m3,
                                                        const int* __restrict__ mask3,
                                                        float* __restrict__ sp0,
                                                        float* __restrict__ sp1,
                                                        int* __restrict__ bar_cnt,
                                                        int* __restrict__ bar_gen) {
  __shared__ float ldsbuf[32768];          // 128 KB: [0,16K)=Wh frags, [16K,32K)=sp frags
  float* whLds = ldsbuf;
  float* spLds = ldsbuf + 16384;
  int lane = threadIdx.x & 31;
  int wid  = threadIdx.x >> 5;
  int wg   = blockIdx.x;                   // 0..15
  int clustered = read_cluster_id() != 0;

  for (int idx = threadIdx.x; idx < 2 * 128 * 32; idx += 128) {
    int lane_i = idx & 31;
    int ks     = (idx >> 5) & 127;
    int nt_i   = idx >> 12;
    int j  = wg * 32 + nt_i * 16 + (lane_i & 15);
    int kk = ks * 4 + (lane_i >> 4) * 2;
    async_copy8(Wh + (size_t)j * FC + kk, whLds + idx * 2);
  }
  __builtin_amdgcn_s_wait_asynccnt(0);
  __syncthreads();

  int mt = wid >> 1, nt = wid & 1;
  int l15 = lane & 15, lh = lane >> 4;
  int jglob = wg * 32 + nt * 16 + l15;
  float alpha = expf(-1.f / tau_m3[jglob]);
  float ro    = expf(-1.f / tau_adp3[jglob]);
  float onea = 1.f - alpha, oner = 1.f - ro;

  float mem[8], spike[8], bb[8];
  #pragma unroll
  for (int v = 0; v < 8; ++v) { mem[v] = 0.f; spike[v] = 0.f; bb[v] = 0.1f; }

  const float* Bb = whLds + (size_t)nt * (128 * 64) + lane * 2;
  const float* Ab = spLds + (size_t)mt * (128 * 64) + lane * 2;

  for (int t = 0; t < TT; ++t) {
    const float* cur = (t & 1) ? sp1 : sp0;
    float*       nxt = (t & 1) ? sp0 : sp1;

    for (int idx = threadIdx.x; idx < 2 * 128 * 32; idx += 128) {
      int lane_i = idx & 31;
      int ks     = (idx >> 5) & 127;
      int mt_i   = idx >> 12;
      int m  = mt_i * 16 + (lane_i & 15);
      int kk = ks * 4 + (lane_i >> 4) * 2;
      async_copy8(cur + (size_t)m * FC + kk, spLds + idx * 2);
    }
    __builtin_amdgcn_s_wait_asynccnt(0);
    __syncthreads();

    v8f acc;
    #pragma unroll
    for (int v = 0; v < 8; ++v)
      acc[v] = G3[(size_t)(t * NB + mt * 16 + v + lh * 8) * FC + jglob];

    #pragma unroll 4
    for (int ks = 0; ks < 128; ++ks) {
      v2f a = *(const v2f*)(Ab + ks * 64);
      v2f b = *(const v2f*)(Bb + ks * 64);
      acc = wmma_f32(a, b, acc);
    }

    int mk = mask3[jglob * TT + t];
    #pragma unroll
    for (int v = 0; v < 8; ++v) {
      bb[v] = ro * bb[v] + oner * spike[v];
      float Bth = 0.1f + 1.8f * bb[v];
      float nm  = mem[v] * alpha + onea * acc[v] - Bth * spike[v];
      if (mk) mem[v] = nm;
      spike[v] = (mk && (mem[v] - Bth > 0.f)) ? 1.f : 0.f;
      int m = mt * 16 + v + lh * 8;
      nxt[(size_t)m * FC + jglob] = spike[v];
    }
    grid_barrier(bar_cnt, bar_gen, t, clustered);
  }
}

// ---------------- readout: out = sp3_last @ Wo^T + bo ----------------
__global__ void out_kernel(const float* __restrict__ sp,
                           const float* __restrict__ Wo,
                           const float* __restrict__ bo,
                           float* __restrict__ out) {
  int tid = blockIdx.x * blockDim.x + threadIdx.x;
  if (tid >= NB * OUTD) return;
  int o = tid % OUTD, n = tid / OUTD;
  float s = bo[o];
  for (int k = 0; k < FC; ++k) s += sp[(size_t)n * FC + k] * Wo[(size_t)o * FC + k];
  out[tid] = s;
}

extern "C" void kernel_launch(void* const* d_in, const int* in_sizes, int n_in,
                              void* d_out, int out_size, void* d_ws, size_t ws_size,
                              hipStream_t stream) {
  const float* x        = (const float*)d_in[0];
  const float* W1       = (const float*)d_in[1];
  const float* b1       = (const float*)d_in[2];
  const float* W2       = (const float*)d_in[3];
  const float* b2       = (const float*)d_in[4];
  const float* W3       = (const float*)d_in[5];
  const float* b3       = (const float*)d_in[6];
  const float* Wh       = (const float*)d_in[7];
  const float* bh       = (const float*)d_in[8];
  const float* Wo       = (const float*)d_in[9];
  const float* bo       = (const float*)d_in[10];
  const float* tau_adp3 = (const float*)d_in[13];
  const float* tau_m1   = (const float*)d_in[14];
  const float* tau_m2   = (const float*)d_in[15];
  const float* tau_m3   = (const float*)d_in[16];
  const int*   mask1    = (const int*)d_in[17];
  const int*   mask2    = (const int*)d_in[18];
  const int*   mask3    = (const int*)d_in[19];
  float* out = (float*)d_out;

  float* w = (float*)d_ws;
  size_t off = 0;
  float* X0    = w + off; off += (size_t)MROWS * IND;
  float* X1    = w + off; off += (size_t)MROWS * IND;
  float* H1    = w + off; off += (size_t)MROWS * FC;
  float* H2    = w + off; off += (size_t)MROWS * FC;
  float* SPcat = w + off; off += (size_t)MROWS * 2 * FC;
  float* G3    = w + off; off += (size_t)MROWS * FC;
  float* sp0   = w + off; off += (size_t)NB * FC;
  float* sp1   = w + off; off += (size_t)NB * FC;
  int*   bar   = (int*)(w + off);

  init_kernel<<<64, 256, 0, stream>>>(sp0, sp1, bar);

  pool_kernel<<<(2 * MROWS * IND) / 256, 256, 0, stream>>>(x, X0, X1);

  dim3 ggrid(FC / 128, MROWS / 64);
  gemm_wmma<<<ggrid, 256, 0, stream>>>(X0, W1, b1, nullptr, H1, MROWS, IND, FC);
  gemm_wmma<<<ggrid, 256, 0, stream>>>(X1, W2, b2, nullptr, H2, MROWS, IND, FC);

  scan12_kernel<<<(2 * NB * FC) / 256, 256, 0, stream>>>(H1, H2, mask1, mask2,
                                                         tau_m1, tau_m2, SPcat);

  gemm_wmma<<<ggrid, 256, 0, stream>>>(SPcat, W3, b3, bh, G3, MROWS, 2 * FC, FC);

  recurrent_kernel<<<16, 128, 0, stream>>>(G3, Wh, tau_adp3, tau_m3, mask3,
                                           sp0, sp1, bar, bar + 1);

  out_kernel<<<(NB * OUTD + 255) / 256, 256, 0, stream>>>(sp0, Wo, bo, out);
}